// EntityGNN_86535001080497
// MI455X (gfx1250) — compile-verified
//
#include <hip/hip_runtime.h>

typedef __attribute__((ext_vector_type(2))) float v2f;
typedef __attribute__((ext_vector_type(8))) float v8f;

#define N_NODES   50000
#define N_EDGES   800000
#define DIM       64
#define N_GRAPHS  64
#define BN_EPS    1e-5f
#define ROW_TILES (N_NODES / 16)          // 3125 exactly
#define N_WAVES   (ROW_TILES * 4)         // 12500 (rowtile x 4 coltiles)

// Guaranteed hardware GLOBAL_ATOMIC_ADD_F32 (no CAS fallback), independent of
// -munsafe-fp-atomics. Non-returning -> tracked with STOREcnt only.
__device__ __forceinline__ void atomAddF32(float* p, float v) {
    unsafeAtomicAdd(p, v);
}

// ---------------------------------------------------------------- init
__global__ void init_kernel(float* __restrict__ deg, float* __restrict__ agg,
                            float* __restrict__ stats, float* __restrict__ pool) {
    unsigned idx = blockIdx.x * blockDim.x + threadIdx.x;
    if (idx < (unsigned)(N_NODES * DIM)) agg[idx] = 0.0f;
    if (idx < N_NODES)                   deg[idx] = 1.0f;   // self-loop contribution
    if (idx < 256)                       stats[idx] = 0.0f; // sums + sumsq
    if (idx < N_GRAPHS * DIM + N_GRAPHS) pool[idx] = 0.0f;  // pool sums + counts
}

__global__ void zero_kernel(float* __restrict__ p, unsigned n) {
    unsigned idx = blockIdx.x * blockDim.x + threadIdx.x;
    if (idx < n) p[idx] = 0.0f;
}

// ---------------------------------------------------------------- degree / dinv
__global__ void degree_kernel(const int* __restrict__ dst, float* __restrict__ deg) {
    unsigned e = blockIdx.x * blockDim.x + threadIdx.x;
    if (e < N_EDGES) atomAddF32(&deg[dst[e]], 1.0f);
}

__global__ void rsqrt_kernel(float* __restrict__ deg) {
    unsigned n = blockIdx.x * blockDim.x + threadIdx.x;
    if (n < N_NODES) {
        float d = deg[n];
        deg[n] = (d > 0.0f) ? rsqrtf(d) : 0.0f;
    }
}

// ---------------------------------------------------------------- f32 WMMA GEMM: C[N,64] = A[N,64] @ W[64,64]
// One wave computes a 16x16 output tile, K=64 as 16 steps of V_WMMA_F32_16X16X4_F32.
__global__ __launch_bounds__(256) void gemm_wmma_kernel(const float* __restrict__ A,
                                                        const float* __restrict__ W,
                                                        float* __restrict__ C) {
    const unsigned wid  = threadIdx.x >> 5;
    const unsigned lane = threadIdx.x & 31;
    const unsigned gw   = blockIdx.x * 8u + wid;
    if (gw >= N_WAVES) return;                 // wave-uniform: EXEC stays all-1 for WMMA
    const unsigned rt = gw >> 2;               // row tile 0..3124
    const unsigned ct = gw & 3;                // col tile 0..3
    const unsigned m16   = lane & 15;
    const unsigned row   = rt * 16 + m16;      // A row for this lane
    const unsigned col   = ct * 16 + m16;      // B/C column for this lane
    const unsigned khalf = (lane >> 4) * 2;    // lanes 0-15 -> K {0,1}; lanes 16-31 -> K {2,3}

    v8f acc = {};
#pragma unroll
    for (unsigned kb = 0; kb < DIM; kb += 4) {
        v2f a, b;
        a.x = A[(size_t)row * DIM + kb + khalf + 0];
        a.y = A[(size_t)row * DIM + kb + khalf + 1];
        b.x = W[(size_t)(kb + khalf + 0) * DIM + col];
        b.y = W[(size_t)(kb + khalf + 1) * DIM + col];
        acc = __builtin_amdgcn_wmma_f32_16x16x4_f32(false, a, false, b,
                                                    (short)0, acc, false, false);
    }
    // C/D layout: VGPR v -> M = v + 8*(lane>=16), N = lane&15
    const unsigned mbase = rt * 16 + ((lane >> 4) << 3);
#pragma unroll
    for (int v = 0; v < 8; ++v) {
        C[(size_t)(mbase + v) * DIM + col] = acc[v];
    }
}

// ---------------------------------------------------------------- edge gather * norm -> atomic scatter
// 16 threads per edge, 4 floats (float4) each. Self-loops appended as edges [E, E+N).
__global__ void scatter_kernel(const float* __restrict__ h,
                               const int* __restrict__ src,
                               const int* __restrict__ dst,
                               const float* __restrict__ dinv,
                               float* __restrict__ out) {
    unsigned tid = blockIdx.x * blockDim.x + threadIdx.x;
    const unsigned total = (unsigned)(N_EDGES + N_NODES) * 16u;
    if (tid >= total) return;
    unsigned e = tid >> 4;
    unsigned q = tid & 15;
    int s, d;
    if (e < N_EDGES) { s = src[e]; d = dst[e]; }
    else             { s = d = (int)(e - N_EDGES); }
    float norm = dinv[s] * dinv[d];
    const float4 v = *reinterpret_cast<const float4*>(h + (size_t)s * DIM + q * 4u);
    float* o = out + (size_t)d * DIM + q * 4u;
    atomAddF32(o + 0, v.x * norm);
    atomAddF32(o + 1, v.y * norm);
    atomAddF32(o + 2, v.z * norm);
    atomAddF32(o + 3, v.w * norm);
}

// ---------------------------------------------------------------- per-column sum / sumsq (BN stats)
__global__ __launch_bounds__(256) void colstats_kernel(const float* __restrict__ g,
                                                       float* __restrict__ stats) {
    __shared__ float ssum[256];
    __shared__ float ssq[256];
    const unsigned c     = threadIdx.x & 63;
    const unsigned rlane = threadIdx.x >> 6;   // 0..3
    float sum = 0.0f, sq = 0.0f;
    for (unsigned row = blockIdx.x * 4u + rlane; row < N_NODES; row += gridDim.x * 4u) {
        float v = g[(size_t)row * DIM + c];
        sum += v; sq += v * v;
    }
    ssum[threadIdx.x] = sum;
    ssq[threadIdx.x]  = sq;
    __syncthreads();
    if (threadIdx.x < 64) {
        sum = ssum[threadIdx.x] + ssum[threadIdx.x + 64] + ssum[threadIdx.x + 128] + ssum[threadIdx.x + 192];
        sq  = ssq[threadIdx.x]  + ssq[threadIdx.x + 64]  + ssq[threadIdx.x + 128]  + ssq[threadIdx.x + 192];
        atomAddF32(&stats[c], sum);
        atomAddF32(&stats[64 + c], sq);
    }
}

// ---------------------------------------------------------------- BN coefficients (b1 cancels in BN)
__global__ void bncoef_kernel(const float* __restrict__ stats,
                              const float* __restrict__ gamma,
                              const float* __restrict__ beta,
                              float* __restrict__ coef) {
    int c = threadIdx.x;
    if (c < 64) {
        const float invN = 1.0f / (float)N_NODES;
        float mu  = stats[c] * invN;
        float var = stats[64 + c] * invN - mu * mu;   // biased var, matches jnp.var
        float a   = gamma[c] * rsqrtf(var + BN_EPS);
        coef[c]      = a;
        coef[64 + c] = beta[c] - mu * a;
    }
}

__global__ void bnrelu_kernel(float* __restrict__ g, const float* __restrict__ coef) {
    unsigned idx = blockIdx.x * blockDim.x + threadIdx.x;
    if (idx < (unsigned)(N_NODES * DIM)) {
        unsigned c = idx & 63;
        g[idx] = fmaxf(fmaf(g[idx], coef[c], coef[64 + c]), 0.0f);
    }
}

// ---------------------------------------------------------------- global mean pool
__global__ void pool_kernel(const float* __restrict__ h, const int* __restrict__ batch,
                            float* __restrict__ pool) {
    unsigned idx = blockIdx.x * blockDim.x + threadIdx.x;
    if (idx < (unsigned)(N_NODES * DIM)) {
        unsigned n = idx >> 6, c = idx & 63;
        int g = batch[n];
        atomAddF32(&pool[(size_t)g * DIM + c], h[idx]);
    }
}

__global__ void count_kernel(const int* __restrict__ batch, float* __restrict__ cnt) {
    unsigned n = blockIdx.x * blockDim.x + threadIdx.x;
    if (n < N_NODES) atomAddF32(&cnt[batch[n]], 1.0f);
}

__global__ void pool_final_kernel(const float* __restrict__ pool, const float* __restrict__ cnt,
                                  const float* __restrict__ b2, float* __restrict__ out) {
    unsigned idx = blockIdx.x * blockDim.x + threadIdx.x;
    if (idx < N_GRAPHS * DIM) {
        unsigned g = idx >> 6, c = idx & 63;
        float k = cnt[g];
        // node-wise +b2 commutes through the mean (k*b2/k); empty graphs -> 0 (sums==0)
        out[idx] = (pool[idx] + k * b2[c]) / fmaxf(k, 1.0f);
    }
}

// ---------------------------------------------------------------- launch
extern "C" void kernel_launch(void* const* d_in, const int* in_sizes, int n_in,
                              void* d_out, int out_size, void* d_ws, size_t ws_size,
                              hipStream_t stream) {
    const float* x     = (const float*)d_in[0];
    const int*   ei    = (const int*)  d_in[1];   // [2, E]
    const int*   batch = (const int*)  d_in[2];   // [N]
    const float* W1    = (const float*)d_in[3];
    /* b1 = d_in[4] : cancels inside BatchNorm (mean over axis 0) */
    const float* gamma = (const float*)d_in[5];
    const float* beta  = (const float*)d_in[6];
    const float* W2    = (const float*)d_in[7];
    const float* b2    = (const float*)d_in[8];
    float*       out   = (float*)d_out;

    float* ws    = (float*)d_ws;
    float* deg   = ws;                                // N (deg -> dinv in place)
    float* bufA  = deg  + N_NODES;                    // N*64 (GEMM outputs)
    float* bufB  = bufA + (size_t)N_NODES * DIM;      // N*64 (aggregation / relu)
    float* stats = bufB + (size_t)N_NODES * DIM;      // 128  (colsum, colsumsq)
    float* coef  = stats + 128;                       // 128  (scale, shift)
    float* pool  = coef  + 128;                       // 64*64 sums
    float* cnt   = pool  + N_GRAPHS * DIM;            // 64 counts (contiguous with pool)

    const int* src = ei;
    const int* dst = ei + N_EDGES;

    const unsigned nd       = (unsigned)(N_NODES * DIM);
    const unsigned total16  = (unsigned)(N_EDGES + N_NODES) * 16u;
    const unsigned gemmGrid = (N_WAVES + 7u) / 8u;    // 8 waves per 256-thread block

    init_kernel  <<<(nd + 255) / 256, 256, 0, stream>>>(deg, bufB, stats, pool);
    degree_kernel<<<(N_EDGES + 255) / 256, 256, 0, stream>>>(dst, deg);
    rsqrt_kernel <<<(N_NODES + 255) / 256, 256, 0, stream>>>(deg);

    // Layer 1: h = x @ W1 (WMMA) -> normalized scatter -> BN + ReLU
    gemm_wmma_kernel<<<gemmGrid, 256, 0, stream>>>(x, W1, bufA);
    scatter_kernel  <<<(total16 + 255) / 256, 256, 0, stream>>>(bufA, src, dst, deg, bufB);
    colstats_kernel <<<256, 256, 0, stream>>>(bufB, stats);
    bncoef_kernel   <<<1, 64, 0, stream>>>(stats, gamma, beta, coef);
    bnrelu_kernel   <<<(nd + 255) / 256, 256, 0, stream>>>(bufB, coef);

    // Layer 2: h2 = relu @ W2 (WMMA) -> normalized scatter
    gemm_wmma_kernel<<<gemmGrid, 256, 0, stream>>>(bufB, W2, bufA);
    zero_kernel     <<<(nd + 255) / 256, 256, 0, stream>>>(bufB, nd);
    scatter_kernel  <<<(total16 + 255) / 256, 256, 0, stream>>>(bufA, src, dst, deg, bufB);

    // Global mean pool (+b2 folded in)
    pool_kernel      <<<(nd + 255) / 256, 256, 0, stream>>>(bufB, batch, pool);
    count_kernel     <<<(N_NODES + 255) / 256, 256, 0, stream>>>(batch, cnt);
    pool_final_kernel<<<(N_GRAPHS * DIM + 255) / 256, 256, 0, stream>>>(pool, cnt, b2, out);
}